// DecoderGravity_33268816675213
// MI455X (gfx1250) — compile-verified
//
#include <hip/hip_runtime.h>
#include <hip/hip_bf16.h>

// Gravity decoder, edge-list branch:
//   out[e] = x[dst[e],128] - l * log( ||x[src[e],0:128] - x[dst[e],0:128]||^2 + 0.01 )
//
// Strategy (MI455X / gfx1250, wave32):
//  - one wave per 16-edge tile
//  - stage 16 diff rows (16 x 128 f32) into LDS with coalesced b32 reads
//  - reduce all 16 squared distances at once with V_WMMA_F32_16X16X4_F32:
//      C(16x16) += A(16x4) * B(4x16) over 32 K-steps, A == B == diff fragment,
//      keep only the diagonal of C (per-edge dot(diff,diff)).
//  - lanes 0-7 / 24-31 hold the diagonal -> epilogue + scattered store.

typedef __attribute__((ext_vector_type(2))) float v2f;
typedef __attribute__((ext_vector_type(8))) float v8f;

#define ROW_STRIDE   129   // floats per node row (128 pos + 1 mass)
#define D_POS        128
#define LDS_STRIDE   132   // padded: banks (4*e + 4*k) % 64 distinct across 16 lanes
#define WAVES_PER_BLOCK 4
#define GRAV_EPS     0.01f

__global__ __launch_bounds__(32 * WAVES_PER_BLOCK)
void DecoderGravity_kernel(const float* __restrict__ x,
                           const int*   __restrict__ eidx,   // [2, E] row-major
                           const float* __restrict__ lparam, // [1]
                           float*       __restrict__ out,    // [E]
                           int E)
{
    __shared__ float tile[WAVES_PER_BLOCK][16][LDS_STRIDE];

    const int lane   = threadIdx.x & 31;
    const int waveIb = threadIdx.x >> 5;
    const int wavesTotal = gridDim.x * WAVES_PER_BLOCK;
    const int T = (E + 15) >> 4;                  // 16-edge tiles
    const float lval = lparam[0];
    const int* __restrict__ srcIdx = eidx;        // row 0
    const int* __restrict__ dstIdx = eidx + E;    // row 1

    float (* __restrict__ my)[LDS_STRIDE] = tile[waveIb];

    for (int t = blockIdx.x * WAVES_PER_BLOCK + waveIb; t < T; t += wavesTotal) {
        const int tileBase = t << 4;

        // ---- stage: diff rows -> LDS, coalesced 128B requests per wave ----
        #pragma unroll 1
        for (int e = 0; e < 16; ++e) {
            const int edge = tileBase + e;
            const int inb  = (edge < E);
            const int sRow = inb ? srcIdx[edge] : 0;   // uniform -> s_load
            const int dRow = inb ? dstIdx[edge] : 0;
            const float* __restrict__ ps = x + (size_t)sRow * ROW_STRIDE;
            const float* __restrict__ pd = x + (size_t)dRow * ROW_STRIDE;
            #pragma unroll
            for (int c = 0; c < 4; ++c) {
                const int dpos = lane + (c << 5);
                my[e][dpos] = ps[dpos] - pd[dpos];
            }
        }
        // LDS slice is private to this wave: same-wave DS ops are in-order,
        // compiler inserts s_wait_dscnt before the fragment reads. No barrier.

        // ---- WMMA reduction: C += A * A^T over K = 0..127 in steps of 4 ----
        // A-fragment (16x4 f32): lanes 0-15 hold row M=lane, K={0,1};
        //                        lanes 16-31 hold row M=lane-16, K={2,3}.
        // B-fragment (4x16) of the transposed source carries the identical
        // per-lane float2, so one register pair feeds both operands.
        const int eL   = lane & 15;
        const int koff = (lane >> 4) << 1;         // 0 or 2
        const float* __restrict__ arow = &my[eL][0];

        v8f acc = {};
        #pragma unroll
        for (int k = 0; k < 32; ++k) {
            v2f a;
            a.x = arow[(k << 2) + koff + 0];
            a.y = arow[(k << 2) + koff + 1];
            acc = __builtin_amdgcn_wmma_f32_16x16x4_f32(
                /*neg_a=*/false, a, /*neg_b=*/false, a,
                /*c_mod=*/(short)0, acc, /*reuse_a=*/false, /*reuse_b=*/false);
        }

        // ---- diagonal extraction (16x16 f32 C/D layout) ----
        // edge m in [0,8):  VGPR m,   lane m
        // edge m in [8,16): VGPR m-8, lane m+16
        int myEdge = -1;
        if (lane < 8)        myEdge = lane;
        else if (lane >= 24) myEdge = lane - 16;

        const int v = lane & 7;
        float r2 = acc[0];
        if (v == 1) r2 = acc[1];
        if (v == 2) r2 = acc[2];
        if (v == 3) r2 = acc[3];
        if (v == 4) r2 = acc[4];
        if (v == 5) r2 = acc[5];
        if (v == 6) r2 = acc[6];
        if (v == 7) r2 = acc[7];

        if (myEdge >= 0) {
            const int edge = tileBase + myEdge;
            if (edge < E) {
                const float mj = x[(size_t)dstIdx[edge] * ROW_STRIDE + D_POS];
                out[edge] = mj - lval * logf(r2 + GRAV_EPS);
            }
        }
    }
}

extern "C" void kernel_launch(void* const* d_in, const int* in_sizes, int n_in,
                              void* d_out, int out_size, void* d_ws, size_t ws_size,
                              hipStream_t stream)
{
    const float* x      = (const float*)d_in[0];   // [N, 129] f32
    const int*   eidx   = (const int*)  d_in[1];   // [2, E]
    const float* lparam = (const float*)d_in[2];   // [1]
    float*       out    = (float*)d_out;           // [E]

    const int E = in_sizes[1] / 2;
    const int T = (E + 15) / 16;                   // 16-edge tiles, 1 per wave
    int blocks = (T + WAVES_PER_BLOCK - 1) / WAVES_PER_BLOCK;
    if (blocks < 1) blocks = 1;

    DecoderGravity_kernel<<<blocks, 32 * WAVES_PER_BLOCK, 0, stream>>>(
        x, eidx, lparam, out, E);
}